// LinearSDEGenerator_76501957476838
// MI455X (gfx1250) — compile-verified
//
#include <hip/hip_runtime.h>

// Linear SDE, 500 Euler-Maruyama steps over 16384 3-D paths, save every 5th.
// One V_WMMA_F32_16X16X4_F32 per 16-path block per step. K=4 carries the
// augmented state [x0,x1,x2,1]. A-matrix rows:
//   rows 0-2 : I + dt*driftT      (col3 = dt*drift_bias)   -> drift-updated x
//   rows 3-5 : sqrt_dt*diffT      (col3 = sqrt_dt*diff_b)  -> pre-scaled g
//   rows 8-13: duplicate of rows 0-5
// Row duplication makes the WMMA broadcast f and g into BOTH 16-lane halves
// (D rows 0-7 -> lanes 0-15, rows 8-15 -> lanes 16-31, same column), so the
// per-step update is pure FMA + cndmask: no shuffles, no LDS, no scratch.

typedef float v2f __attribute__((ext_vector_type(2)));
typedef float v8f __attribute__((ext_vector_type(8)));

#define B_PATHS    16384
#define N_STEPS    500
#define SAVE_EVERY 5
#define N_SAVES    100
#define T_SAVED    101          // x0 + 100 saves
#define DT         0.002f
#define PF_NEAR    8            // WGP-scope prefetch distance (steps)
#define PF_FAR     32           // SE-scope  prefetch distance (steps)

__global__ __launch_bounds__(64) void sde_wmma_kernel(
    const float* __restrict__ drift, const float* __restrict__ drift_bias,
    const float* __restrict__ diffusion, const float* __restrict__ diffusion_bias,
    const float* __restrict__ x0, const float* __restrict__ noise,
    float* __restrict__ out)
{
    const int  lane = threadIdx.x & 31;
    const int  wave = (blockIdx.x * blockDim.x + threadIdx.x) >> 5;
    const int  base = wave * 32;          // 32 paths per wave (2 WMMA blocks)
    const int  m    = lane & 15;          // A-row (this lane's half) / path idx
    const bool lo   = lane < 16;
    const float sqdt = __builtin_sqrtf(DT);

    // ---------------- Build A (16x4 f32, 2 VGPRs/lane) ----------------
    // lane<16 holds A[m][0],A[m][1]; lane>=16 holds A[m][2],A[m][3].
    // Rows 8-13 duplicate rows 0-5  =>  mm = m & 7 selects the content.
    const int k0 = lo ? 0 : 2;
    const int k1 = lo ? 1 : 3;
    const int mm = m & 7;
    float a0v = 0.0f, a1v = 0.0f;
    if (mm < 3) {                               // drift-aug rows
        a0v = DT * drift[mm * 3 + k0] + ((mm == k0) ? 1.0f : 0.0f);
        a1v = (k1 < 3) ? (DT * drift[mm * 3 + k1] + ((mm == k1) ? 1.0f : 0.0f))
                       : (DT * drift_bias[mm]);
    } else if (mm < 6) {                        // scaled diffusion rows
        const int r = mm - 3;
        a0v = sqdt * diffusion[r * 3 + k0];
        a1v = (k1 < 3) ? (sqdt * diffusion[r * 3 + k1])
                       : (sqdt * diffusion_bias[r]);
    }
    v2f Amat; Amat.x = a0v; Amat.y = a1v;

    // ---------------- Initial state in B layout (4x16 augmented) -----------
    // B v0: lanes<16 = K0 row (x0), lanes>=16 = K2 row (x2)
    // B v1: lanes<16 = K1 row (x1), lanes>=16 = K3 row (1.0)
    const int p0 = base + m;          // block0 path (both halves)
    const int p1 = base + 16 + m;     // block1 path (both halves)
    v2f B0, B1;
    B0.x = lo ? x0[p0 * 3 + 0] : x0[p0 * 3 + 2];
    B0.y = lo ? x0[p0 * 3 + 1] : 1.0f;
    B1.x = lo ? x0[p1 * 3 + 0] : x0[p1 * 3 + 2];
    B1.y = lo ? x0[p1 * 3 + 1] : 1.0f;

    // ---------------- Write t=0 (x0) ----------------
    const int p  = base + lane;                 // each path handled by one lane
    const int BT = B_PATHS * T_SAVED;
    out[0 * BT + p * T_SAVED] = x0[p * 3 + 0];
    out[1 * BT + p * T_SAVED] = x0[p * 3 + 1];
    out[2 * BT + p * T_SAVED] = x0[p * 3 + 2];

    // ---------------- Time loop: 100 x (5 steps + 1 save) ----------------
    for (int s = 0; s < N_SAVES; ++s) {
        float sv0 = 0.0f, sv1 = 0.0f, sv2 = 0.0f;
#pragma unroll
        for (int k = 0; k < SAVE_EVERY; ++k) {
            const int t   = s * SAVE_EVERY + k;
            const int off = t * (B_PATHS * 3);

            // Near prefetch: WGP scope (all cache levels). 32 lanes cover the
            // wave's full 384B noise slab for step t+PF_NEAR.
            int tn = t + PF_NEAR; if (tn >= N_STEPS) tn = N_STEPS - 1;
            const float* pfn = noise + tn * (B_PATHS * 3) + p * 3;
            asm volatile("global_prefetch_b8 %0, off" :: "v"(pfn));
            // Far prefetch: SE scope (HBM -> GL2) further ahead.
            int tf = t + PF_FAR; if (tf >= N_STEPS) tf = N_STEPS - 1;
            __builtin_prefetch(noise + tf * (B_PATHS * 3) + p * 3, 0, 3);

            // Per-step noise (lanes l and l+16 read the same 12B line).
            const float n00 = noise[off + p0 * 3 + 0];
            const float n01 = noise[off + p0 * 3 + 1];
            const float n02 = noise[off + p0 * 3 + 2];
            const float n10 = noise[off + p1 * 3 + 0];
            const float n11 = noise[off + p1 * 3 + 1];
            const float n12 = noise[off + p1 * 3 + 2];

            v8f cz = {};
            v8f d0 = __builtin_amdgcn_wmma_f32_16x16x4_f32(
                false, Amat, false, B0, (short)0, cz, false, false);
            v8f d1 = __builtin_amdgcn_wmma_f32_16x16x4_f32(
                false, Amat, false, B1, (short)0, cz, false, false);

            // d[r] = drift-updated x_r, d[3+r] = sqdt*g_r, identical in both
            // halves (duplicated A rows). x_new = f + g*dW, all 32 lanes.
            const float xn00 = __builtin_fmaf(d0[3], n00, d0[0]);
            const float xn01 = __builtin_fmaf(d0[4], n01, d0[1]);
            const float xn02 = __builtin_fmaf(d0[5], n02, d0[2]);
            const float xn10 = __builtin_fmaf(d1[3], n10, d1[0]);
            const float xn11 = __builtin_fmaf(d1[4], n11, d1[1]);
            const float xn12 = __builtin_fmaf(d1[5], n12, d1[2]);

            // Rebuild B operands: pure cndmask, no cross-lane movement.
            B0.x = lo ? xn00 : xn02;  B0.y = lo ? xn01 : 1.0f;
            B1.x = lo ? xn10 : xn12;  B1.y = lo ? xn11 : 1.0f;

            if (k == SAVE_EVERY - 1) {          // compile-time branch
                sv0 = lo ? xn00 : xn10;         // lane l>=16 carries block1,
                sv1 = lo ? xn01 : xn11;         // path base+l, matching p
                sv2 = lo ? xn02 : xn12;
            }
        }
        const int so = p * T_SAVED + (s + 1);
        out[0 * BT + so] = sv0;
        out[1 * BT + so] = sv1;
        out[2 * BT + so] = sv2;
    }
}

extern "C" void kernel_launch(void* const* d_in, const int* in_sizes, int n_in,
                              void* d_out, int out_size, void* d_ws, size_t ws_size,
                              hipStream_t stream) {
    const float* drift     = (const float*)d_in[0];
    const float* driftb    = (const float*)d_in[1];
    const float* diffusion = (const float*)d_in[2];
    const float* diffb     = (const float*)d_in[3];
    const float* x0        = (const float*)d_in[4];
    const float* noise     = (const float*)d_in[5];
    float* out = (float*)d_out;

    dim3 block(64);                        // 2 wave32s per block
    dim3 grid(B_PATHS / 64);               // 256 blocks -> spread across WGPs
    sde_wmma_kernel<<<grid, block, 0, stream>>>(
        drift, driftb, diffusion, diffb, x0, noise, out);
}